// GAT_47536698032658
// MI455X (gfx1250) — compile-verified
//
#include <hip/hip_runtime.h>
#include <math.h>

typedef float v2f __attribute__((ext_vector_type(2)));
typedef float v8f __attribute__((ext_vector_type(8)));

#define F_IN   128
#define F_OUT  32
#define HEADS  8
#define NTYPE  16
#define C_COLS (F_OUT * HEADS)   /* 256 projected columns, j = f*8 + h */

// ---------------------------------------------------------------------------
// Kernel A: nf = x @ fc_w  via V_WMMA_F32_16X16X4_F32 (full fp32 precision).
// One 16x16 C tile per wave32; 8 waves per 256-thread block.
// A 16x4 f32 layout (ISA 7.12.2): lanes 0-15 hold K={0,1} in v0/v1 for row M=lane;
// lanes 16-31 hold K={2,3}. B 4x16 mirrors it (lane = column). C/D: VGPR r holds
// row r (lanes 0-15) / row r+8 (lanes 16-31), lane&15 = column.
// ---------------------------------------------------------------------------
__global__ __launch_bounds__(256) void gat_gemm_wmma(
    const float* __restrict__ X, const float* __restrict__ W,
    float* __restrict__ NF, int nrows, int totalTiles)
{
  const int wave   = threadIdx.x >> 5;
  const int lane   = threadIdx.x & 31;
  const int tileId = blockIdx.x * 8 + wave;
  if (tileId >= totalTiles) return;            // wave-uniform exit
  const int tm = tileId / (C_COLS / 16);
  const int tn = tileId % (C_COLS / 16);

  const int r15 = lane & 15;
  const int kp  = (lane >> 4) << 1;            // 0 or 2

  int arow = tm * 16 + r15;                    // clamp via select: no divergence
  if (arow >= nrows) arow = nrows - 1;
  const float* Xr  = X + (size_t)arow * F_IN;
  const int    bcol = tn * 16 + r15;

  v8f c = {};
  for (int k = 0; k < F_IN; k += 4) {
    v2f a, b;
    a.x = Xr[k + kp];
    a.y = Xr[k + kp + 1];
    b.x = W[(size_t)(k + kp    ) * C_COLS + bcol];
    b.y = W[(size_t)(k + kp + 1) * C_COLS + bcol];
    c = __builtin_amdgcn_wmma_f32_16x16x4_f32(
            /*neg_a=*/false, a, /*neg_b=*/false, b,
            /*c_mod=*/(short)0, c, /*reuse_a=*/false, /*reuse_b=*/false);
  }

  const int crow0 = tm * 16 + ((lane >> 4) << 3);
#pragma unroll
  for (int r = 0; r < 8; ++r) {
    const int row = crow0 + r;
    if (row < nrows) NF[(size_t)row * C_COLS + bcol] = c[r];
  }
}

// ---------------------------------------------------------------------------
// Kernel B: a1[n,h] = sum_f nf[n,f,h]*attn_l[f,h]
//           ntype[n,t,h] = sum_f nf[n,f,h]*attn_r[t,f,h]   (params in LDS)
// ---------------------------------------------------------------------------
__global__ __launch_bounds__(256) void gat_node_logits(
    const float* __restrict__ NF, const float* __restrict__ AL,
    const float* __restrict__ AR, float* __restrict__ A1,
    float* __restrict__ NT, int n)
{
  __shared__ float sL[F_OUT * HEADS];            // 256 floats
  __shared__ float sR[NTYPE * F_OUT * HEADS];    // 4096 floats
  sL[threadIdx.x] = AL[threadIdx.x];
  for (int i = threadIdx.x; i < NTYPE * F_OUT * HEADS; i += 256) sR[i] = AR[i];
  __syncthreads();

  const int gid = blockIdx.x * 256 + threadIdx.x;
  if (gid >= n * HEADS) return;
  const int node = gid >> 3, h = gid & 7;
  const float* row = NF + (size_t)node * C_COLS;

  float acc1 = 0.f;
  float acct[NTYPE];
#pragma unroll
  for (int t = 0; t < NTYPE; ++t) acct[t] = 0.f;

  for (int f = 0; f < F_OUT; ++f) {
    const float v = row[f * HEADS + h];
    acc1 = fmaf(v, sL[f * HEADS + h], acc1);
#pragma unroll
    for (int t = 0; t < NTYPE; ++t)
      acct[t] = fmaf(v, sR[t * (F_OUT * HEADS) + f * HEADS + h], acct[t]);
  }
  A1[gid] = acc1;
  float* nt = NT + (size_t)node * (NTYPE * HEADS);
#pragma unroll
  for (int t = 0; t < NTYPE; ++t) nt[t * HEADS + h] = acct[t];
}

// Monotone float<->uint mapping so segment-max can use integer atomicMax
// (order-independent => deterministic). Init value 0 sits below flip(-inf).
__device__ __forceinline__ unsigned f2mono(float f) {
  const unsigned u = __float_as_uint(f);
  return (u & 0x80000000u) ? ~u : (u | 0x80000000u);
}
__device__ __forceinline__ float mono2f(unsigned m) {
  return (m & 0x80000000u) ? __uint_as_float(m & 0x7FFFFFFFu)
                           : __uint_as_float(~m);
}

// ---------------------------------------------------------------------------
// Kernel C1: per-(edge,head) logit + leaky_relu; segment max into mmax.
// ---------------------------------------------------------------------------
__global__ __launch_bounds__(256) void gat_edge_logit_max(
    const float* __restrict__ A1, const float* __restrict__ NT,
    const int* __restrict__ src, const int* __restrict__ dst,
    const int* __restrict__ et, float* __restrict__ aout,
    unsigned* __restrict__ mmax, int e)
{
  const int gid = blockIdx.x * 256 + threadIdx.x;
  if (gid >= e * HEADS) return;
  const int ed = gid >> 3, h = gid & 7;
  const int s = src[ed], d = dst[ed], t = et[ed];
  float v = A1[d * HEADS + h] + NT[(size_t)s * (NTYPE * HEADS) + t * HEADS + h];
  v = (v > 0.f) ? v : 0.2f * v;                 // leaky_relu slope 0.2
  aout[gid] = v;
  atomicMax(&mmax[d * HEADS + h], f2mono(v));
}

// ---------------------------------------------------------------------------
// Kernel C2: ex = exp(a - m[dst]); segment sum into den. (a buffer reused)
// ---------------------------------------------------------------------------
__global__ __launch_bounds__(256) void gat_edge_exp_sum(
    const int* __restrict__ dst, float* __restrict__ aout,
    const unsigned* __restrict__ mmax, float* __restrict__ den, int e)
{
  const int gid = blockIdx.x * 256 + threadIdx.x;
  if (gid >= e * HEADS) return;
  const int ed = gid >> 3, h = gid & 7;
  const int d = dst[ed];
  const float m  = mono2f(mmax[d * HEADS + h]);
  const float ex = expf(aout[gid] - m);
  aout[gid] = ex;
  atomicAdd(&den[d * HEADS + h], ex);
}

// ---------------------------------------------------------------------------
// Kernel C3: attn = ex/den[dst] -> d_out; scatter-add attn * nf[src] into hbuf.
// hbuf (51 MB) + nf (51 MB) are L2-resident on the 192 MB L2.
// ---------------------------------------------------------------------------
__global__ __launch_bounds__(256) void gat_edge_norm_scatter(
    const float* __restrict__ NF, const int* __restrict__ src,
    const int* __restrict__ dst, float* __restrict__ aout,
    const float* __restrict__ den, float* __restrict__ hbuf, int e)
{
  const int gid = blockIdx.x * 256 + threadIdx.x;
  if (gid >= e * HEADS) return;
  const int ed = gid >> 3, h = gid & 7;
  const int s = src[ed], d = dst[ed];
  const float attn = aout[gid] / den[d * HEADS + h];
  aout[gid] = attn;
  const float* srow = NF + (size_t)s * C_COLS;
  float*       drow = hbuf + (size_t)d * C_COLS;
#pragma unroll
  for (int f = 0; f < F_OUT; ++f)
    atomicAdd(&drow[f * HEADS + h], srow[f * HEADS + h] * attn);
}

// ---------------------------------------------------------------------------
// Kernel D: ret[n,f] = elu( (1/8) * sum_h2 (hbuf[n, h2*32+f] + bias[h2*32+f]) )
// (the reference's reshape trick: storage order is j=f*8+h, read j=h2*32+f)
// ---------------------------------------------------------------------------
__global__ __launch_bounds__(256) void gat_finalize(
    const float* __restrict__ hbuf, const float* __restrict__ bias,
    float* __restrict__ ret, int n)
{
  const int gid = blockIdx.x * 256 + threadIdx.x;
  if (gid >= n * F_OUT) return;
  const int node = gid >> 5, f = gid & 31;
  const float* row = hbuf + (size_t)node * C_COLS;
  float acc = 0.f;
#pragma unroll
  for (int h2 = 0; h2 < HEADS; ++h2) {
    const int j = h2 * F_OUT + f;
    acc += row[j] + bias[j];
  }
  const float v = acc * (1.0f / HEADS);
  ret[gid] = (v > 0.f) ? v : expm1f(v);         // elu, alpha=1
}

// ---------------------------------------------------------------------------
extern "C" void kernel_launch(void* const* d_in, const int* in_sizes, int n_in,
                              void* d_out, int out_size, void* d_ws, size_t ws_size,
                              hipStream_t stream)
{
  const float* x      = (const float*)d_in[0];
  const float* fc_w   = (const float*)d_in[1];
  const float* attn_l = (const float*)d_in[2];
  const float* attn_r = (const float*)d_in[3];
  const float* bias   = (const float*)d_in[4];
  const int*   src    = (const int*)d_in[5];
  const int*   dst    = (const int*)d_in[6];
  const int*   etype  = (const int*)d_in[7];

  const int n = in_sizes[0] / F_IN;   // 50000
  const int e = in_sizes[5];          // 800000

  // Workspace layout (floats): nf | a1 | ntype | mmax(u32) | den | hbuf
  float*    nf    = (float*)d_ws;
  float*    a1    = nf    + (size_t)n * C_COLS;
  float*    ntype = a1    + (size_t)n * HEADS;
  unsigned* mmax  = (unsigned*)(ntype + (size_t)n * NTYPE * HEADS);
  float*    den   = (float*)(mmax + (size_t)n * HEADS);
  float*    hbuf  = den   + (size_t)n * HEADS;

  float* attn_out = (float*)d_out;                    // [E,1,H] flat, h fastest
  float* ret_out  = attn_out + (size_t)e * HEADS;     // [N, F_OUT]

  // Zero mmax|den|hbuf in one contiguous async memset (graph-capture safe).
  hipMemsetAsync(mmax, 0, (size_t)n * (2 * HEADS + C_COLS) * sizeof(float), stream);

  const int tmCount    = (n + 15) / 16;
  const int totalTiles = tmCount * (C_COLS / 16);
  gat_gemm_wmma<<<(totalTiles + 7) / 8, 256, 0, stream>>>(x, fc_w, nf, n, totalTiles);

  const int nh = n * HEADS;
  gat_node_logits<<<(nh + 255) / 256, 256, 0, stream>>>(nf, attn_l, attn_r, a1, ntype, n);

  const int eh = e * HEADS;
  gat_edge_logit_max   <<<(eh + 255) / 256, 256, 0, stream>>>(a1, ntype, src, dst, etype,
                                                              attn_out, mmax, e);
  gat_edge_exp_sum     <<<(eh + 255) / 256, 256, 0, stream>>>(dst, attn_out, mmax, den, e);
  gat_edge_norm_scatter<<<(eh + 255) / 256, 256, 0, stream>>>(nf, src, dst, attn_out, den,
                                                              hbuf, e);
  const int nf2 = n * F_OUT;
  gat_finalize<<<(nf2 + 255) / 256, 256, 0, stream>>>(hbuf, bias, ret_out, n);
}